// GraphEncoder_12661563588730
// MI455X (gfx1250) — compile-verified
//
#include <hip/hip_runtime.h>

#define N_NODES 50000
#define N_EDGES 800000
#define N_GRAPHS 64
#define NODE_F 11
#define EDGE_F 14
#define H 128
#define PDIM 128
#define NLAYERS 3
#define BN_EPS 1e-5f
#define SCAN_BLOCKS ((N_NODES + 255) / 256)   // 196

typedef __attribute__((ext_vector_type(2))) float v2f;
typedef __attribute__((ext_vector_type(8))) float v8f;

// ---------------------------------------------------------------- utilities
__global__ void zero_f32(float* __restrict__ p, int n) {
    int i = blockIdx.x * blockDim.x + threadIdx.x;
    if (i < n) p[i] = 0.0f;
}
__global__ void zero_i32(int* __restrict__ p, int n) {
    int i = blockIdx.x * blockDim.x + threadIdx.x;
    if (i < n) p[i] = 0;
}

// ------------------------------------------------- node embedding: h = x @ node_W^T + b
__global__ void node_embed(const float* __restrict__ x, const float* __restrict__ W,
                           const float* __restrict__ b, float* __restrict__ h) {
    int t = blockIdx.x * blockDim.x + threadIdx.x;
    if (t >= N_NODES * H) return;
    int n = t >> 7, f = t & 127;
    const float* xr = x + (long long)n * NODE_F;
    const float* wr = W + f * NODE_F;
    float acc = b[f];
#pragma unroll
    for (int k = 0; k < NODE_F; ++k) acc += xr[k] * wr[k];
    h[t] = acc;
}

// ------------------------ counting sort of edges by dst (built once, reused 3 layers)
__global__ void edge_hist(const int* __restrict__ eidx, int* __restrict__ deg) {
    int e = blockIdx.x * blockDim.x + threadIdx.x;
    if (e < N_EDGES) atomicAdd(deg + eidx[N_EDGES + e], 1);
}

__global__ void scan_blocks(const int* __restrict__ deg, int* __restrict__ base,
                            int* __restrict__ parts) {
    __shared__ int buf[256];
    int g = blockIdx.x * 256 + threadIdx.x;
    int v = (g < N_NODES) ? deg[g] : 0;
    buf[threadIdx.x] = v;
    __syncthreads();
#pragma unroll
    for (int off = 1; off < 256; off <<= 1) {
        int t = (threadIdx.x >= off) ? buf[threadIdx.x - off] : 0;
        __syncthreads();
        buf[threadIdx.x] += t;
        __syncthreads();
    }
    if (g < N_NODES) base[g] = buf[threadIdx.x] - v;   // exclusive within block
    if (threadIdx.x == 255) parts[blockIdx.x] = buf[255];
}

__global__ void scan_parts(int* __restrict__ parts, int nparts) {
    __shared__ int buf[256];
    int v = (threadIdx.x < nparts) ? parts[threadIdx.x] : 0;
    buf[threadIdx.x] = v;
    __syncthreads();
#pragma unroll
    for (int off = 1; off < 256; off <<= 1) {
        int t = (threadIdx.x >= off) ? buf[threadIdx.x - off] : 0;
        __syncthreads();
        buf[threadIdx.x] += t;
        __syncthreads();
    }
    if (threadIdx.x < nparts) parts[threadIdx.x] = buf[threadIdx.x] - v;  // exclusive
}

__global__ void scan_add(int* __restrict__ base, const int* __restrict__ parts) {
    int g = blockIdx.x * 256 + threadIdx.x;
    if (g < N_NODES) base[g] += parts[blockIdx.x];
}

__global__ void scatter_perm(const int* __restrict__ eidx, const int* __restrict__ base,
                             int* __restrict__ cursor, int* __restrict__ perm) {
    int e = blockIdx.x * blockDim.x + threadIdx.x;
    if (e >= N_EDGES) return;
    int dst = eidx[N_EDGES + e];
    int r = atomicAdd(cursor + dst, 1);
    perm[base[dst] + r] = e;
}

// ---- atomic-free aggregation: agg[n] = sum_{e: dst(e)=n} relu(h[src(e)] + edgeMLP(e))
// One wave per node; lane owns 4 features; edge list is contiguous after the sort.
// h[src] gathers are coalesced 512B rows and L2-resident (h = 25.6 MB << 192 MB L2).
__global__ __launch_bounds__(256) void gather_agg(
        const float* __restrict__ h, const int* __restrict__ eidx,
        const float* __restrict__ eattr, const float* __restrict__ eW,
        const float* __restrict__ eb, const int* __restrict__ base,
        const int* __restrict__ perm, float* __restrict__ agg) {
    int node = blockIdx.x * 8 + (threadIdx.x >> 5);
    if (node >= N_NODES) return;
    int lane = threadIdx.x & 31;
    int f0 = lane * 4;
    int s = base[node];
    int epos_end = (node == N_NODES - 1) ? N_EDGES : base[node + 1];

    float w[4][EDGE_F];
    float ebv[4];
#pragma unroll
    for (int j = 0; j < 4; ++j) {
        ebv[j] = eb[f0 + j];
#pragma unroll
        for (int k = 0; k < EDGE_F; ++k) w[j][k] = eW[(f0 + j) * EDGE_F + k];
    }
    float acc[4] = {0.f, 0.f, 0.f, 0.f};
    for (int p = s; p < epos_end; ++p) {
        int eid = perm[p];
        int src = eidx[eid];
        const float* er = eattr + (long long)eid * EDGE_F;
        float ea[EDGE_F];
#pragma unroll
        for (int k = 0; k < EDGE_F; ++k) ea[k] = er[k];
        float4 hv = *(const float4*)(h + (long long)src * H + f0);
        float hs[4] = {hv.x, hv.y, hv.z, hv.w};
#pragma unroll
        for (int j = 0; j < 4; ++j) {
            float ef = ebv[j];
#pragma unroll
            for (int k = 0; k < EDGE_F; ++k) ef += ea[k] * w[j][k];
            float m = hs[j] + ef;
            acc[j] += (m > 0.0f) ? m : 0.0f;
        }
    }
    float4 o = {acc[0], acc[1], acc[2], acc[3]};
    *(float4*)(agg + (long long)node * H + f0) = o;
}

// ---- fused layer: h = relu( BNReLU((h+agg) @ W1^T + b1) @ W2^T + b2 ) + h
// Block = 256 threads = 8 waves; block owns 16 rows; wave w owns N-tile [16w,16w+16).
// A and z1 live in LDS (row stride 132 floats -> conflict-free, 64 banks).
// GEMMs via V_WMMA_F32_16X16X4_F32 (exact f32 semantics), 32 WMMA/wave/GEMM.
__global__ __launch_bounds__(256) void gine_layer(
        float* __restrict__ h, const float* __restrict__ agg,
        const float* __restrict__ W1, const float* __restrict__ b1,
        const float* __restrict__ gamma, const float* __restrict__ beta,
        const float* __restrict__ mean, const float* __restrict__ var,
        const float* __restrict__ W2, const float* __restrict__ b2) {
    __shared__ float Abuf[16 * 132];
    __shared__ float Zbuf[16 * 132];
    const int tid = threadIdx.x;
    const int m0  = blockIdx.x * 16;

    {   // stage A = h + agg (16x128)
        int row = tid >> 4;
        int c0  = (tid & 15) * 8;
        const float* hr = h   + (long long)(m0 + row) * H + c0;
        const float* ar = agg + (long long)(m0 + row) * H + c0;
#pragma unroll
        for (int j = 0; j < 8; ++j) Abuf[row * 132 + c0 + j] = hr[j] + ar[j];
    }
    __syncthreads();

    const int lane  = tid & 31;
    const int wave  = tid >> 5;
    const int n0    = wave * 16;
    const int cl    = lane & 15;
    const int koff  = (lane >> 4) * 2;
    const int rbase = (lane >> 4) << 3;

    // ---------------- GEMM1 ----------------
    v8f acc;
    {
        float bias = b1[n0 + cl];
#pragma unroll
        for (int v = 0; v < 8; ++v) acc[v] = bias;
    }
    const float* w1r = W1 + (long long)(n0 + cl) * H;
#pragma unroll 4
    for (int kb = 0; kb < H; kb += 4) {
        v2f a, b;
        a.x = Abuf[cl * 132 + kb + koff];
        a.y = Abuf[cl * 132 + kb + koff + 1];
        b.x = w1r[kb + koff];
        b.y = w1r[kb + koff + 1];
        acc = __builtin_amdgcn_wmma_f32_16x16x4_f32(false, a, false, b,
                                                    (short)0, acc, false, false);
    }
    {   // BN (eval) + ReLU -> Zbuf
        int col  = n0 + cl;
        float sc = gamma[col] * rsqrtf(var[col] + BN_EPS);
        float mu = mean[col];
        float bt = beta[col];
#pragma unroll
        for (int v = 0; v < 8; ++v) {
            int row = v + rbase;
            float z = (acc[v] - mu) * sc + bt;
            z = z > 0.0f ? z : 0.0f;
            Zbuf[row * 132 + col] = z;
        }
    }
    __syncthreads();

    // ---------------- GEMM2 ----------------
    v8f acc2;
    {
        float bias = b2[n0 + cl];
#pragma unroll
        for (int v = 0; v < 8; ++v) acc2[v] = bias;
    }
    const float* w2r = W2 + (long long)(n0 + cl) * H;
#pragma unroll 4
    for (int kb = 0; kb < H; kb += 4) {
        v2f a, b;
        a.x = Zbuf[cl * 132 + kb + koff];
        a.y = Zbuf[cl * 132 + kb + koff + 1];
        b.x = w2r[kb + koff];
        b.y = w2r[kb + koff + 1];
        acc2 = __builtin_amdgcn_wmma_f32_16x16x4_f32(false, a, false, b,
                                                     (short)0, acc2, false, false);
    }
#pragma unroll
    for (int v = 0; v < 8; ++v) {   // ReLU + residual, in-place on own rows
        int row = v + rbase;
        long long idx = (long long)(m0 + row) * H + n0 + cl;
        float z = acc2[v];
        z = z > 0.0f ? z : 0.0f;
        h[idx] = z + h[idx];
    }
}

// ---------------------------------------------------------------- pooling
__global__ void pool_count(const int* __restrict__ bidx, float* __restrict__ counts) {
    int n = blockIdx.x * blockDim.x + threadIdx.x;
    if (n < N_NODES) atomicAdd(counts + bidx[n], 1.0f);
}

__global__ void pool_sum(const float* __restrict__ h, const int* __restrict__ bidx,
                         float* __restrict__ pooled) {
    int t = blockIdx.x * blockDim.x + threadIdx.x;
    if (t >= N_NODES * H) return;
    int n = t >> 7, f = t & 127;
    atomicAdd(pooled + bidx[n] * H + f, h[t]);
}

__global__ void final_proj(const float* __restrict__ pooled, const float* __restrict__ counts,
                           const float* __restrict__ pW, const float* __restrict__ pb,
                           float* __restrict__ out) {
    int t = blockIdx.x * blockDim.x + threadIdx.x;
    if (t >= N_GRAPHS * PDIM) return;
    int g = t >> 7, p = t & 127;
    float inv = 1.0f / fmaxf(counts[g], 1.0f);
    const float* pr = pooled + g * H;
    const float* wr = pW + p * H;
    float acc = pb[p];
#pragma unroll 8
    for (int k = 0; k < H; ++k) acc += (pr[k] * inv) * wr[k];
    out[t] = acc;
}

// ---------------------------------------------------------------- launch
extern "C" void kernel_launch(void* const* d_in, const int* in_sizes, int n_in,
                              void* d_out, int out_size, void* d_ws, size_t ws_size,
                              hipStream_t stream) {
    const float* x      = (const float*)d_in[0];
    const int*   eidx   = (const int*)  d_in[1];
    const float* eattr  = (const float*)d_in[2];
    const int*   bidx   = (const int*)  d_in[3];
    const float* node_W = (const float*)d_in[4];
    const float* node_b = (const float*)d_in[5];
    const float* edge_W = (const float*)d_in[6];
    const float* edge_b = (const float*)d_in[7];
    const float* lin1_W = (const float*)d_in[8];
    const float* lin1_b = (const float*)d_in[9];
    const float* bn_g   = (const float*)d_in[10];
    const float* bn_b   = (const float*)d_in[11];
    const float* bn_m   = (const float*)d_in[12];
    const float* bn_v   = (const float*)d_in[13];
    const float* lin2_W = (const float*)d_in[14];
    const float* lin2_b = (const float*)d_in[15];
    const float* proj_W = (const float*)d_in[16];
    const float* proj_b = (const float*)d_in[17];
    float* out = (float*)d_out;

    // ws: h[N*H] | agg[N*H] | pooled[G*H] | counts[G] | deg[N] | base[N] | perm[E] | parts[256]
    float* h      = (float*)d_ws;
    float* agg    = h + (size_t)N_NODES * H;
    float* pooled = agg + (size_t)N_NODES * H;
    float* counts = pooled + (size_t)N_GRAPHS * H;
    int*   deg    = (int*)(counts + N_GRAPHS);    // also reused as scatter cursor
    int*   base   = deg + N_NODES;
    int*   perm   = base + N_NODES;
    int*   parts  = perm + N_EDGES;

    // ---- counting sort of edges by dst (once; reused for all layers) ----
    zero_i32<<<(N_NODES + 255) / 256, 256, 0, stream>>>(deg, N_NODES);
    edge_hist<<<(N_EDGES + 255) / 256, 256, 0, stream>>>(eidx, deg);
    scan_blocks<<<SCAN_BLOCKS, 256, 0, stream>>>(deg, base, parts);
    scan_parts<<<1, 256, 0, stream>>>(parts, SCAN_BLOCKS);
    scan_add<<<SCAN_BLOCKS, 256, 0, stream>>>(base, parts);
    zero_i32<<<(N_NODES + 255) / 256, 256, 0, stream>>>(deg, N_NODES);  // cursor
    scatter_perm<<<(N_EDGES + 255) / 256, 256, 0, stream>>>(eidx, base, deg, perm);

    // ---- node embedding ----
    {
        int n = N_NODES * H;
        node_embed<<<(n + 255) / 256, 256, 0, stream>>>(x, node_W, node_b, h);
    }
    // ---- layers ----
    for (int i = 0; i < NLAYERS; ++i) {
        gather_agg<<<(N_NODES + 7) / 8, 256, 0, stream>>>(h, eidx, eattr, edge_W, edge_b,
                                                          base, perm, agg);
        gine_layer<<<N_NODES / 16, 256, 0, stream>>>(
            h, agg,
            lin1_W + (size_t)i * H * H, lin1_b + i * H,
            bn_g + i * H, bn_b + i * H, bn_m + i * H, bn_v + i * H,
            lin2_W + (size_t)i * H * H, lin2_b + i * H);
    }
    // ---- pooling + projection ----
    {
        int n = N_GRAPHS * H + N_GRAPHS;   // pooled and counts are contiguous
        zero_f32<<<(n + 255) / 256, 256, 0, stream>>>(pooled, n);
        pool_count<<<(N_NODES + 255) / 256, 256, 0, stream>>>(bidx, counts);
        int nt = N_NODES * H;
        pool_sum<<<(nt + 255) / 256, 256, 0, stream>>>(h, bidx, pooled);
        final_proj<<<(N_GRAPHS * PDIM + 255) / 256, 256, 0, stream>>>(pooled, counts,
                                                                      proj_W, proj_b, out);
    }
}